// utilFace_11201274708633
// MI455X (gfx1250) — compile-verified
//
#include <hip/hip_runtime.h>
#include <math.h>

#define B_N 512
#define E_K 512
#define C_N 70722
#define TILE_M 128
#define TILE_N 128
#define TILE_K 16
#define NTHREADS 256
#define NSLAB (E_K / TILE_K)
#define AS_STRIDE 17    // 16 + 1 pad dword  (TDM pad_interval=3, pad_amount=0)
#define BS_STRIDE 136   // 128 + 8 pad dwords (TDM pad_interval=6, pad_amount=7)

static constexpr float F_M   = 0.4f;
static constexpr float F_H   = 0.333f;
static constexpr float F_S   = 64.0f;
static constexpr float F_EPS = 0.001f;
static constexpr float PI_F  = 3.14159265358979323846f;

typedef __attribute__((ext_vector_type(2))) float v2f;
typedef __attribute__((ext_vector_type(8))) float v8f;
typedef unsigned int u32;
typedef unsigned long long u64;
typedef u32 __attribute__((ext_vector_type(4))) v4u;
typedef int __attribute__((ext_vector_type(4))) v4i;
typedef int __attribute__((ext_vector_type(8))) v8i;

// ---------------------------------------------------------------------------
// LDS byte offset of a __shared__ object (generic -> AS3 -> int).
// ---------------------------------------------------------------------------
__device__ __forceinline__ u32 lds_addr_of(const void* p) {
    return (u32)(u64)(const __attribute__((address_space(3))) void*)p;
}

// ---------------------------------------------------------------------------
// Issue one 2D TDM load: tile (tile_d0 x tile_d1) of 4-byte elements from a
// row-major tensor with row stride d0_stride, into LDS at lds_byte, with
// optional LDS row padding.  tensor_d0/d1 = remaining extent from tile origin
// (OOB elements are zero-filled / suppressed).  Tracked by TENSORcnt.
// ---------------------------------------------------------------------------
__device__ __forceinline__ void tdm_load_2d(const void* gaddr, u32 lds_byte,
                                            u32 tensor_d0, u32 tensor_d1,
                                            u64 d0_stride,
                                            u32 tile_d0, u32 tile_d1,
                                            u32 pad_en, u32 pad_int, u32 pad_amt) {
    const u64 ga = (u64)gaddr;
    v4u g0;
    g0.x = 1u;                                            // count=1, user descriptor
    g0.y = lds_byte;                                      // lds_addr
    g0.z = (u32)(ga & 0xFFFFFFFFu);                       // global_addr[31:0]
    g0.w = (u32)((ga >> 32) & 0x1FFFFFFu) | (2u << 30);   // global_addr[56:32] | type=2
    v8i g1;
    g1[0] = (int)((2u << 16) | (pad_en << 20) | (pad_int << 22) | (pad_amt << 25));
    g1[1] = (int)((tensor_d0 & 0xFFFFu) << 16);                       // dim0[15:0]
    g1[2] = (int)((tensor_d0 >> 16) | ((tensor_d1 & 0xFFFFu) << 16)); // dim0[31:16]|dim1[15:0]
    g1[3] = (int)((tensor_d1 >> 16) | (tile_d0 << 16));               // dim1[31:16]|tile0
    g1[4] = (int)(tile_d1 & 0xFFFFu);                                 // tile1 | tile2=0
    g1[5] = (int)(u32)(d0_stride & 0xFFFFFFFFu);
    g1[6] = (int)(u32)((d0_stride >> 32) & 0xFFFFu);                  // stride0[47:32]
    g1[7] = 0;
    const v4i z4 = {0, 0, 0, 0};
#if __clang_major__ >= 23
    const v8i z8 = {0, 0, 0, 0, 0, 0, 0, 0};
    __builtin_amdgcn_tensor_load_to_lds(g0, g1, z4, z4, z8, 0);
#else
    __builtin_amdgcn_tensor_load_to_lds(g0, g1, z4, z4, 0);
#endif
}

// ---------------------------------------------------------------------------
// Kernel 1: batch stats, group means, margin scalers, KL scalar.
// Single block of 512 threads (16 waves).
// ---------------------------------------------------------------------------
__global__ void stats_kernel(const float* __restrict__ norms,
                             const int* __restrict__ label,
                             float* __restrict__ out,
                             float* __restrict__ ws_ms,
                             float* __restrict__ ws_cw) {
    __shared__ float s_sn[B_N];
    __shared__ int   s_lb[B_N];
    __shared__ float red[B_N];

    const int t = threadIdx.x;
    float sn = fminf(fmaxf(norms[t], 0.001f), 100.0f);
    s_sn[t] = sn;
    s_lb[t] = label[t];
    __syncthreads();

    red[t] = sn; __syncthreads();
    for (int s = B_N / 2; s > 0; s >>= 1) { if (t < s) red[t] += red[t + s]; __syncthreads(); }
    const float total = red[0];
    __syncthreads();
    const float mean = total * (1.0f / (float)B_N);

    red[t] = sn * sn; __syncthreads();
    for (int s = B_N / 2; s > 0; s >>= 1) { if (t < s) red[t] += red[t + s]; __syncthreads(); }
    const float sumsq = red[0];
    __syncthreads();
    const float var   = fmaxf((sumsq - (float)B_N * mean * mean) / (float)(B_N - 1), 0.0f);
    const float denom = sqrtf(var) + F_EPS;

    float gs = 0.f, gc = 0.f;
    const int ml = s_lb[t];
    for (int j = 0; j < B_N; ++j)
        if (s_lb[j] == ml) { gs += s_sn[j]; gc += 1.0f; }
    const float gm = gs / fmaxf(gc, 1.0f);

    float ms = fminf(fmaxf((sn - mean) / denom * F_H, -1.0f), 1.0f);
    float cw = fminf(fmaxf((sn - gm)   / denom,       -1.0f), 1.0f) * F_H;
    ws_ms[t] = ms;
    ws_cw[t] = cw;
    out[(size_t)B_N * C_N + t] = cw;                    // output #2

    const float u = 1.0f / (float)B_N;
    float kl = u * (logf(u) - sn / total);
    red[t] = kl; __syncthreads();
    for (int s = B_N / 2; s > 0; s >>= 1) { if (t < s) red[t] += red[t + s]; __syncthreads(); }
    const float sumkl = red[0];
    __syncthreads();
    red[t] = (sn <= mean) ? 1.0f : 0.0f; __syncthreads();
    for (int s = B_N / 2; s > 0; s >>= 1) { if (t < s) red[t] += red[t + s]; __syncthreads(); }
    if (t == 0)
        out[(size_t)B_N * C_N + B_N] = red[0] * sumkl / ((float)B_N * (float)B_N);  // output #3
}

// ---------------------------------------------------------------------------
// Kernel 2: fp32 WMMA GEMM with TDM double-buffered staging and fused
// column-norm + clip/scale epilogue (non-temporal stores keep B hot in L2).
// Block = 256 threads (8 waves), block tile 128x128, wave tile 32x64.
// ---------------------------------------------------------------------------
__global__ __launch_bounds__(NTHREADS)
void gemm_kernel(const float* __restrict__ emb,
                 const float* __restrict__ kern,
                 float* __restrict__ out1,
                 float* __restrict__ out4) {
    __shared__ float As[2][TILE_M * AS_STRIDE];   // TDM-padded: row r at r*17
    __shared__ float Bs[2][TILE_K * BS_STRIDE];   // TDM-padded: row k at k*136
    __shared__ float inv_lds[TILE_N];

    const int tid  = threadIdx.x;
    const int m0   = blockIdx.x * TILE_M;         // M fast -> L2 reuse of B across M-blocks
    const int n0   = blockIdx.y * TILE_N;
    const int lane = tid & 31;
    const int half = lane >> 4;                   // fp32 WMMA: lanes 16-31 hold K={2,3}
    const int lr   = lane & 15;
    const int wave = tid >> 5;
    const int mBase = (wave & 3) * 32;
    const int nBase = (wave >> 2) * 64;

    v8f acc[2][4];
    #pragma unroll
    for (int i = 0; i < 2; ++i)
        #pragma unroll
        for (int j = 0; j < 4; ++j)
            acc[i][j] = (v8f){0.f,0.f,0.f,0.f,0.f,0.f,0.f,0.f};

    float ss = 0.f;                               // per-column sum of squares (threads<128)

    // Prologue: stage slab 0 into buffer 0 (wave 0 issues; EXEC ignored by TDM).
    if (tid < 32) {
        tdm_load_2d(emb + (size_t)m0 * E_K, lds_addr_of(&As[0][0]),
                    E_K, (u32)(B_N - m0), E_K, TILE_K, TILE_M, 1u, 3u, 0u);
        tdm_load_2d(kern + n0, lds_addr_of(&Bs[0][0]),
                    (u32)(C_N - n0), E_K, C_N, TILE_N, TILE_K, 1u, 6u, 7u);
    }

    for (int s = 0; s < NSLAB; ++s) {
        const int buf = s & 1;
        if (tid < 32) {
            if (s + 1 < NSLAB) {                  // stage next slab into other buffer
                const int kt = (s + 1) * TILE_K;
                tdm_load_2d(emb + (size_t)m0 * E_K + kt, lds_addr_of(&As[buf ^ 1][0]),
                            (u32)(E_K - kt), (u32)(B_N - m0), E_K, TILE_K, TILE_M, 1u, 3u, 0u);
                tdm_load_2d(kern + (size_t)kt * C_N + n0, lds_addr_of(&Bs[buf ^ 1][0]),
                            (u32)(C_N - n0), (u32)(E_K - kt), C_N, TILE_N, TILE_K, 1u, 6u, 7u);
                __builtin_amdgcn_s_wait_tensorcnt(2);   // in-order: slab s retired
            } else {
                __builtin_amdgcn_s_wait_tensorcnt(0);
            }
        }
        __syncthreads();                          // slab s visible to all waves

        #pragma unroll
        for (int kk = 0; kk < TILE_K; kk += 4) {
            v2f a[2], b[4];
            #pragma unroll
            for (int mi = 0; mi < 2; ++mi) {
                const int r = (mBase + mi * 16 + lr) * AS_STRIDE + kk + 2 * half;
                a[mi].x = As[buf][r];
                a[mi].y = As[buf][r + 1];
            }
            #pragma unroll
            for (int ni = 0; ni < 4; ++ni) {
                const int cc = nBase + ni * 16 + lr;
                b[ni].x = Bs[buf][(kk + 2 * half) * BS_STRIDE + cc];
                b[ni].y = Bs[buf][(kk + 2 * half + 1) * BS_STRIDE + cc];
            }
            #pragma unroll
            for (int mi = 0; mi < 2; ++mi)
                #pragma unroll
                for (int ni = 0; ni < 4; ++ni)
                    acc[mi][ni] = __builtin_amdgcn_wmma_f32_16x16x4_f32(
                        false, a[mi], false, b[ni], (short)0, acc[mi][ni], false, false);
        }

        // Fused column sum-of-squares from the staged B tile (no extra HBM pass).
        if (tid < TILE_N) {
            #pragma unroll
            for (int k = 0; k < TILE_K; ++k) {
                const float v = Bs[buf][k * BS_STRIDE + tid];
                ss += v * v;
            }
        }
        __syncthreads();                          // all reads done before TDM overwrites
    }

    if (tid < TILE_N) inv_lds[tid] = rsqrtf(ss);
    __syncthreads();

    // Epilogue: C/D layout — VGPR j = row j (lanes 0-15) / j+8 (lanes 16-31), col = lr.
    // Non-temporal stores: outputs are write-once streams; keep B resident in L2.
    #pragma unroll
    for (int ni = 0; ni < 4; ++ni) {
        const int cl  = nBase + ni * 16 + lr;
        const int col = n0 + cl;
        if (col >= C_N) continue;
        const float inv = inv_lds[cl];
        #pragma unroll
        for (int mi = 0; mi < 2; ++mi) {
            const int rowb = m0 + mBase + mi * 16 + half * 8;
            #pragma unroll
            for (int j = 0; j < 8; ++j) {
                float v = acc[mi][ni][j] * inv;
                v = fminf(fmaxf(v, -1.0f + F_EPS), 1.0f - F_EPS) * F_S;
                const size_t o = (size_t)(rowb + j) * C_N + col;
                __builtin_nontemporal_store(v, out1 + o);
                __builtin_nontemporal_store(v, out4 + o);
            }
        }
    }
}

// ---------------------------------------------------------------------------
// Kernel 3: fix the 512 target entries (b, label[b]) with the margin formula.
// ---------------------------------------------------------------------------
__global__ void fixup_kernel(const int* __restrict__ label,
                             const float* __restrict__ ws_ms,
                             const float* __restrict__ ws_cw,
                             float* __restrict__ out1,
                             float* __restrict__ out4) {
    int b = blockIdx.x * blockDim.x + threadIdx.x;
    if (b >= B_N) return;
    const int c = label[b];
    const size_t idx = (size_t)b * C_N + c;
    const float cosine = out1[idx] * (1.0f / F_S);   // already clipped
    const float theta  = acosf(cosine);
    const float ms = ws_ms[b];
    const float cw = ws_cw[b];
    float tm = fminf(fmaxf(theta - F_M * ms, F_EPS), PI_F - F_EPS);
    out1[idx] = (cosf(tm) - (F_M + F_M * ms)) * F_S;
    float tc = fminf(fmaxf(theta - F_M * cw, F_EPS), PI_F - F_EPS);
    out4[idx] = (cosf(tc) - (F_M + F_M * cw)) * F_S;
}

// ---------------------------------------------------------------------------
extern "C" void kernel_launch(void* const* d_in, const int* in_sizes, int n_in,
                              void* d_out, int out_size, void* d_ws, size_t ws_size,
                              hipStream_t stream) {
    const float* emb   = (const float*)d_in[0];   // (512,512)
    const float* norms = (const float*)d_in[1];   // (512,1)
    const int*   label = (const int*)d_in[2];     // (512,)
    const float* kern  = (const float*)d_in[3];   // (512,70722)

    float* out  = (float*)d_out;
    float* out1 = out;
    float* out4 = out + (size_t)B_N * C_N + B_N + 1;

    float* ws_ms = (float*)d_ws;                  // B floats
    float* ws_cw = ws_ms + B_N;                   // B floats

    stats_kernel<<<1, B_N, 0, stream>>>(norms, label, out, ws_ms, ws_cw);

    dim3 grid((B_N + TILE_M - 1) / TILE_M, (C_N + TILE_N - 1) / TILE_N);
    gemm_kernel<<<grid, NTHREADS, 0, stream>>>(emb, kern, out1, out4);

    fixup_kernel<<<2, 256, 0, stream>>>(label, ws_ms, ws_cw, out1, out4);
}